// LissomCortexLayer_16011638079790
// MI455X (gfx1250) — compile-verified
//
#include <hip/hip_runtime.h>
#include <hip/hip_bf16.h>

#define SIDE  96
#define NN    (SIDE * SIDE)   // 9216 neurons
#define BATCH 32

typedef __attribute__((ext_vector_type(2))) float v2f;
typedef __attribute__((ext_vector_type(8))) float v8f;

__device__ __forceinline__ float csig(float x) {
    // custom_sigmoid with THETA=0: clamp to [0, 1]
    x = fmaxf(x, 0.0f);
    return fminf(x, 1.0f);
}

// Dense GEMM panel: D[32 x 16-tile] += A[32 x K] * W[K x N] restricted to columns
// [n0, n0+16), using V_WMMA_F32_16X16X4_F32 (full f32 precision; workload is
// HBM-bandwidth bound so f32 WMMA is the right precision).
//
// A layout (16x4 f32): lane = M + 16*(K>=2), vgpr = K&1  -> one b64 load per lane.
// B layout (4x16 f32): lane = N + 16*(K>=2), vgpr = K&1  -> two N-strided b32 loads.
__device__ __forceinline__ void gemm_tile(const float* __restrict__ A,
                                          const float* __restrict__ W,
                                          int n0, int l15, int khalf,
                                          v8f& acc0, v8f& acc1) {
    const float* arow0 = A + (size_t)l15 * NN;         // M-tile 0: rows 0..15
    const float* arow1 = A + (size_t)(l15 + 16) * NN;  // M-tile 1: rows 16..31
    const float* wcol  = W + n0 + l15;                 // this lane's N column
    #pragma unroll 4
    for (int k0 = 0; k0 < NN; k0 += 4) {
        const int k = k0 + 2 * khalf;  // this lane's K pair
        v2f a0 = *(const v2f*)(arow0 + k);
        v2f a1 = *(const v2f*)(arow1 + k);
        v2f b;
        b.x = wcol[(size_t)k * NN];
        b.y = wcol[(size_t)(k + 1) * NN];
        acc0 = __builtin_amdgcn_wmma_f32_16x16x4_f32(false, a0, false, b,
                                                     (short)0, acc0, false, false);
        acc1 = __builtin_amdgcn_wmma_f32_16x16x4_f32(false, a1, false, b,
                                                     (short)0, acc1, false, false);
    }
}

// Stage 1: afferent = clamp(x @ Wr, 0, 1)   -> d_ws
// C/D layout (16x16 f32): lane = N + 16*(M>=8), vgpr = M&7.
__global__ __launch_bounds__(256) void lissom_afferent(const float* __restrict__ X,
                                                       const float* __restrict__ Wr,
                                                       float* __restrict__ aff) {
    const int lane  = threadIdx.x & 31;
    const int wave  = threadIdx.x >> 5;
    const int n0    = (blockIdx.x * 8 + wave) * 16;
    const int l15   = lane & 15;
    const int khalf = lane >> 4;

    v8f acc0 = {};
    v8f acc1 = {};
    gemm_tile(X, Wr, n0, l15, khalf, acc0, acc1);

    const int j = n0 + l15;
    #pragma unroll
    for (int v = 0; v < 8; ++v) {
        const int m = v + 8 * khalf;
        aff[(size_t)m * NN + j]        = csig(acc0[v]);
        aff[(size_t)(m + 16) * NN + j] = csig(acc1[v]);
    }
}

// Stage 2 (fused): inhibitory via dense WMMA GEMM over Wi; excitatory via a
// 13-tap radius-2 stencil (We is circular-masked: dense GEMM would waste
// 340 MB of HBM traffic for <=13 nonzeros per column); then final combine.
__global__ __launch_bounds__(256) void lissom_stage2(const float* __restrict__ aff,
                                                     const float* __restrict__ We,
                                                     const float* __restrict__ Wi,
                                                     float* __restrict__ out) {
    const int lane  = threadIdx.x & 31;
    const int wave  = threadIdx.x >> 5;
    const int n0    = (blockIdx.x * 8 + wave) * 16;
    const int l15   = lane & 15;
    const int khalf = lane >> 4;

    v8f inh0 = {};
    v8f inh1 = {};
    gemm_tile(aff, Wi, n0, l15, khalf, inh0, inh1);

    // --- excitatory stencil: offsets with dx^2 + dy^2 <= 4 (13 taps) ---
    const int j  = n0 + l15;          // 16 | 96, so a tile never straddles a row
    const int jx = j / SIDE;
    const int jy = j - jx * SIDE;
    const int dx[13] = {-2,-1,-1,-1, 0, 0, 0, 0, 0, 1, 1, 1, 2};
    const int dy[13] = { 0,-1, 0, 1,-2,-1, 0, 1, 2,-1, 0, 1, 0};
    float e0[8] = {0, 0, 0, 0, 0, 0, 0, 0};
    float e1[8] = {0, 0, 0, 0, 0, 0, 0, 0};
    #pragma unroll
    for (int t = 0; t < 13; ++t) {
        const int ix = jx + dx[t];
        const int iy = jy + dy[t];
        if ((unsigned)ix < (unsigned)SIDE && (unsigned)iy < (unsigned)SIDE) {
            const int i   = ix * SIDE + iy;
            const float w = We[(size_t)i * NN + j];
            const float* ap = aff + i;
            #pragma unroll
            for (int v = 0; v < 8; ++v) {
                const int m = v + 8 * khalf;
                e0[v] = fmaf(ap[(size_t)m * NN], w, e0[v]);
                e1[v] = fmaf(ap[(size_t)(m + 16) * NN], w, e1[v]);
            }
        }
    }

    // --- combine: clamp(aff + 0.2*clamp(exc) - 0.4*clamp(inh)) ---
    #pragma unroll
    for (int v = 0; v < 8; ++v) {
        const int m0 = v + 8 * khalf;
        const int m1 = m0 + 16;
        const float a0 = aff[(size_t)m0 * NN + j];
        const float a1 = aff[(size_t)m1 * NN + j];
        out[(size_t)m0 * NN + j] = csig(a0 + 0.2f * csig(e0[v]) - 0.4f * csig(inh0[v]));
        out[(size_t)m1 * NN + j] = csig(a1 + 0.2f * csig(e1[v]) - 0.4f * csig(inh1[v]));
    }
}

extern "C" void kernel_launch(void* const* d_in, const int* in_sizes, int n_in,
                              void* d_out, int out_size, void* d_ws, size_t ws_size,
                              hipStream_t stream) {
    (void)in_sizes; (void)n_in; (void)out_size; (void)ws_size;
    const float* X  = (const float*)d_in[0];  // [32, 9216]
    const float* Wr = (const float*)d_in[1];  // [9216, 9216] dense
    const float* We = (const float*)d_in[2];  // [9216, 9216] radius-2 sparse
    const float* Wi = (const float*)d_in[3];  // [9216, 9216] dense
    float* out = (float*)d_out;               // [32, 9216]
    float* aff = (float*)d_ws;                // scratch: 32*9216 f32 = 1.125 MiB

    const int blocks = NN / (16 * 8);  // 576 N-tiles / 8 waves per block = 72
    lissom_afferent<<<blocks, 256, 0, stream>>>(X, Wr, aff);
    lissom_stage2<<<blocks, 256, 0, stream>>>(aff, We, Wi, out);
}